// graphEncoder_52819507806690
// MI455X (gfx1250) — compile-verified
//
#include <hip/hip_runtime.h>
#include <hip/hip_bf16.h>
#include <math.h>

typedef float v2f __attribute__((ext_vector_type(2)));
typedef float v8f __attribute__((ext_vector_type(8)));
typedef int   v4i __attribute__((ext_vector_type(4)));

#define ACT_NONE 0
#define ACT_TANH 1
#define ACT_SIG  2

// problem constants
#define CB   32
#define CV   256
#define CE   4
#define CMF  16
#define CU1  128
#define CU2  64
#define CAUX 128
#define CL1  256
#define CL2  256
#define CEMB 266240
#define BV   (CB*CV)   /* 8192 */

// ---- async global->LDS support (guarded: falls back to plain copies) ------
#if __has_builtin(__builtin_amdgcn_global_load_async_to_lds_b128)
#define HAVE_ASYNC_LDS 1
#else
#define HAVE_ASYNC_LDS 0
#endif

__device__ __forceinline__ void wait_async_zero() {
#if HAVE_ASYNC_LDS
#if __has_builtin(__builtin_amdgcn_s_wait_asynccnt)
  __builtin_amdgcn_s_wait_asynccnt(0);
#else
  asm volatile("s_wait_asynccnt 0x0" ::: "memory");
#endif
#endif
}

__device__ __forceinline__ void cp_g2lds_b128(const float* src, float* dst) {
#if HAVE_ASYNC_LDS
  __builtin_amdgcn_global_load_async_to_lds_b128(
      (__attribute__((address_space(1))) v4i*)const_cast<float*>(src),
      (__attribute__((address_space(3))) v4i*)dst, 0, 0);
#else
  *(float4*)dst = *(const float4*)src;
#endif
}

__device__ __forceinline__ float actf(float v, int act) {
  if (act == ACT_TANH) return tanhf(v);
  if (act == ACT_SIG)  return 1.0f / (1.0f + __expf(-v));
  return v;
}

// ---------------------------------------------------------------------------
// Generic z-batched WMMA fp32 GEMM: C[z][m][n] = act( A[z]@W[z] + bias[z] )
// One wave per 16x16 tile, v_wmma_f32_16x16x4_f32 chained over K.
// ---------------------------------------------------------------------------
__global__ void gemm_wmma_f32(const float* __restrict__ A, int lda, long aZ,
                              const float* __restrict__ W, int ldw, long wZ,
                              const float* __restrict__ bias, long bZ,
                              float* __restrict__ C, int ldc, long cZ, int colOff,
                              int K, int act)
{
  const int z = blockIdx.z;
  A    += (long)z * aZ;
  W    += (long)z * wZ;
  bias += (long)z * bZ;
  C    += (long)z * cZ;

  const int n0   = blockIdx.x * 16;
  const int m0   = blockIdx.y * 16;
  const int lane = threadIdx.x & 31;
  const int col  = lane & 15;   // A-row index AND B/D-column index for this lane
  const int kh   = lane >> 4;

  const float* arow = A + (long)(m0 + col) * lda + 2 * kh;
  const float* bcol = W + (long)(2 * kh) * ldw + n0 + col;

  v8f acc = {};
  for (int k0 = 0; k0 < K; k0 += 4) {
    v2f a = *(const v2f*)(arow + k0);
    v2f b;
    b.x = bcol[(long)k0 * ldw];
    b.y = bcol[(long)(k0 + 1) * ldw];
    acc = __builtin_amdgcn_wmma_f32_16x16x4_f32(false, a, false, b,
                                                (short)0, acc, false, false);
  }

  const float bv = bias[n0 + col];
  #pragma unroll
  for (int i = 0; i < 8; i++) {
    const int m = m0 + i + 8 * kh;
    C[(long)m * ldc + colOff + n0 + col] = actf(acc[i] + bv, act);
  }
}

// ---------------------------------------------------------------------------
// GCL adjacency contraction with LDS-staged A tiles:
//   h[b][v][u] = tanh( sum_e sum_w adj[b][e][v][w] * tmp[e][b*V+w][u]
//                      + selft[b*V+v][u] )  -> ann[:, colOff:colOff+U]
// Block = (b, v-tile); U/16 waves cover all u-tiles and SHARE the adj slab.
// The 16x64 adj chunk is double-buffered in LDS via async global->LDS b128
// copies (ASYNCcnt), padded to a 68-dword row stride so the WMMA A-fragment
// ds_load_b64 pattern (row=lane&15, pair at w+2*(lane>>4)) is bank-conflict
// free (4*row mod 64 injective).  B fragments stream from the 192MB L2.
// ---------------------------------------------------------------------------
#define WCH     64                 /* w-chunk per stage            */
#define ASTRIDE (WCH + 4)          /* padded LDS row stride (dwords)*/
#define NCHUNK  (CE * (CV / WCH))  /* 16 chunks                     */

__global__ void gcl_adj_wmma(const float* __restrict__ adj,   // (B, E+1, V, V)
                             const float* __restrict__ tmp,   // (E, B*V, U)
                             const float* __restrict__ selft, // (B*V, U)
                             float* __restrict__ ann, int ldc, int colOff,
                             int U)
{
  __shared__ __align__(16) float Ash[2][16 * ASTRIDE];

  const int tid  = threadIdx.x;
  const int wave = tid >> 5;           // u-tile owned by this wave
  const int lane = tid & 31;
  const int n0   = wave * 16;
  const int v0   = blockIdx.y * 16;
  const int b    = blockIdx.z;
  const int Bn   = gridDim.z;
  const int col  = lane & 15;
  const int kh   = lane >> 4;
  const int nthr = blockDim.x;

  const long tmpE = (long)Bn * CV * U; // z-stride of tmp over e

  // stage chunk c (e = c>>2, w0 = (c&3)*WCH) into buffer p
  auto issue_copy = [&](int c, int p) {
    const int e   = c >> 2;
    const int w0c = (c & 3) * WCH;
    const float* base = adj + ((long)(b * (CE + 1) + e) * CV + v0) * CV + w0c;
    for (int t = tid; t < 16 * (WCH / 4); t += nthr) {
      const int row = t >> 4;          // 16 b128 segments per row
      const int wq  = (t & 15) * 4;
      cp_g2lds_b128(base + (long)row * CV + wq, &Ash[p][row * ASTRIDE + wq]);
    }
  };

  issue_copy(0, 0);

  v8f acc = {};
  for (int c = 0; c < NCHUNK; c++) {
    wait_async_zero();
    __syncthreads();                   // buffer c&1 fully staged, prior reads done
    if (c + 1 < NCHUNK) {
      issue_copy(c + 1, (c + 1) & 1);
      // warm next B chunk in L2 (global_prefetch_b8)
      const int e2 = (c + 1) >> 2, w02 = ((c + 1) & 3) * WCH;
      __builtin_prefetch(tmp + (long)e2 * tmpE + ((long)b * CV + w02) * U + n0, 0, 1);
    }

    const int e   = c >> 2;
    const int w0c = (c & 3) * WCH;
    const float* bcol = tmp + (long)e * tmpE + ((long)b * CV + w0c + 2 * kh) * U
                            + n0 + col;
    const float* ap = &Ash[c & 1][col * ASTRIDE + 2 * kh];
    for (int k = 0; k < WCH; k += 4) {
      v2f a = *(const v2f*)(ap + k);   // ds_load_b64, conflict-free
      v2f bb;
      bb.x = bcol[(long)k * U];
      bb.y = bcol[(long)(k + 1) * U];
      acc = __builtin_amdgcn_wmma_f32_16x16x4_f32(false, a, false, bb,
                                                  (short)0, acc, false, false);
    }
    __syncthreads();                   // done reading buffer c&1
  }

  #pragma unroll
  for (int i = 0; i < 8; i++) {
    const int m    = v0 + i + 8 * kh;      // v
    const long row = (long)b * CV + m;     // b*V + v
    const float s  = selft[row * U + n0 + col];
    ann[row * ldc + colOff + n0 + col] = tanhf(acc[i] + s);
  }
}

// scatter node features into the concat layouts: ann1[:, :16], ann2[:, 64:80]
__global__ void scatter_node(const float* __restrict__ node,
                             float* __restrict__ ann1, float* __restrict__ ann2)
{
  const int i = blockIdx.x * 256 + threadIdx.x;
  if (i >= BV * CMF) return;
  const int f  = i & (CMF - 1);
  const int bv = i >> 4;
  const float v = node[i];
  ann1[(long)bv * (CMF + CU1) + f] = v;
  ann2[(long)bv * (CU2 + CMF) + CU2 + f] = v;
}

// o1[b][x] = tanh( sum_v i[b,v,x] * j[b,v,x] )
__global__ void reduce_o(const float* __restrict__ ipre,
                         const float* __restrict__ jpre,
                         float* __restrict__ o1)
{
  const int b = blockIdx.x;
  const int x = threadIdx.x;   // 0..127
  float s = 0.f;
  for (int v = 0; v < CV; v++) {
    const long idx = ((long)b * CV + v) * CAUX + x;
    s += ipre[idx] * jpre[idx];
  }
  o1[b * CAUX + x] = tanhf(s);
}

// ---------------------------------------------------------------------------
// Final VAE head: HBM-bound stream of Wmu/Wlv (545MB) with NT hints.
// One output column per thread, 64 fp32 accumulators over the 32-row batch,
// o3 staged transposed in LDS ([k][b]) -> 8 ds b128 broadcasts + 64 FMAs
// per 2 coalesced NT weight dwords.
// ---------------------------------------------------------------------------
__global__ void __launch_bounds__(256)
final_vae(const float* __restrict__ o3,           // (32,256)
          const float* __restrict__ Wmu, const float* __restrict__ bmu,
          const float* __restrict__ Wlv, const float* __restrict__ blv,
          const float* __restrict__ eps,
          float* __restrict__ out)
{
  __shared__ __align__(16) float osh[CL2 * CB];   // 32 KB, [k][b]
  const int tid = threadIdx.x;
  for (int i = tid; i < CB * CL2; i += 256) {
    const int bb = i >> 8;      // CL2 == 256
    const int kk = i & 255;
    osh[kk * CB + bb] = o3[i];
  }
  __syncthreads();

  const long n = (long)blockIdx.x * 256 + tid;

  float accm[CB], accl[CB];
  #pragma unroll
  for (int b = 0; b < CB; b++) { accm[b] = 0.f; accl[b] = 0.f; }

  const float* pm = Wmu + n;
  const float* pl = Wlv + n;
  for (int k = 0; k < CL2; k++) {
    const float wm = __builtin_nontemporal_load(pm); pm += CEMB;
    const float wl = __builtin_nontemporal_load(pl); pl += CEMB;
    const float4* op = (const float4*)&osh[k * CB];
    #pragma unroll
    for (int q = 0; q < 8; q++) {
      const float4 o4 = op[q];
      accm[4*q+0] = fmaf(o4.x, wm, accm[4*q+0]); accl[4*q+0] = fmaf(o4.x, wl, accl[4*q+0]);
      accm[4*q+1] = fmaf(o4.y, wm, accm[4*q+1]); accl[4*q+1] = fmaf(o4.y, wl, accl[4*q+1]);
      accm[4*q+2] = fmaf(o4.z, wm, accm[4*q+2]); accl[4*q+2] = fmaf(o4.z, wl, accl[4*q+2]);
      accm[4*q+3] = fmaf(o4.w, wm, accm[4*q+3]); accl[4*q+3] = fmaf(o4.w, wl, accl[4*q+3]);
    }
  }

  const float bm = bmu[n], bl = blv[n];
  #pragma unroll
  for (int b = 0; b < CB; b++) {
    const long o  = (long)b * CEMB + n;
    const float mu = accm[b] + bm;
    const float lv = accl[b] + bl;
    __builtin_nontemporal_store(mu, out + (long)CB * CEMB + o);       // h_mu
    __builtin_nontemporal_store(lv, out + 2L * CB * CEMB + o);        // h_logvar
    const float e = __builtin_nontemporal_load(eps + o);
    __builtin_nontemporal_store(fmaf(e, __expf(0.5f * lv), mu), out + o); // h_out
  }
}

// ---------------------------------------------------------------------------
extern "C" void kernel_launch(void* const* d_in, const int* in_sizes, int n_in,
                              void* d_out, int out_size, void* d_ws, size_t ws_size,
                              hipStream_t stream)
{
  (void)in_sizes; (void)n_in; (void)out_size; (void)ws_size;

  const float* adj    = (const float*)d_in[0];
  const float* node   = (const float*)d_in[1];
  const float* W_adj0 = (const float*)d_in[2];
  const float* b_adj0 = (const float*)d_in[3];
  const float* W2_0   = (const float*)d_in[4];
  const float* b2_0   = (const float*)d_in[5];
  const float* W_adj1 = (const float*)d_in[6];
  const float* b_adj1 = (const float*)d_in[7];
  const float* W2_1   = (const float*)d_in[8];
  const float* b2_1   = (const float*)d_in[9];
  const float* Wi     = (const float*)d_in[10];
  const float* bi     = (const float*)d_in[11];
  const float* Wj     = (const float*)d_in[12];
  const float* bj     = (const float*)d_in[13];
  const float* Wd1    = (const float*)d_in[14];
  const float* bd1    = (const float*)d_in[15];
  const float* Wd2    = (const float*)d_in[16];
  const float* bd2    = (const float*)d_in[17];
  const float* Wmu    = (const float*)d_in[18];
  const float* bmu    = (const float*)d_in[19];
  const float* Wlv    = (const float*)d_in[20];
  const float* blv    = (const float*)d_in[21];
  const float* eps    = (const float*)d_in[22];
  float* out = (float*)d_out;

  // workspace layout (floats): ~9.2M floats = ~37 MB
  float* ws    = (float*)d_ws;
  float* tmp   = ws;                                  // E*BV*U1 (reused by layer1)
  float* selft = tmp   + (long)CE * BV * CU1;         // BV*U1 (reused)
  float* ann1  = selft + (long)BV * CU1;              // BV*144
  float* ann2  = ann1  + (long)BV * (CMF + CU1);      // BV*80
  float* ipre  = ann2  + (long)BV * (CU2 + CMF);      // BV*128
  float* jpre  = ipre  + (long)BV * CAUX;             // BV*128
  float* o1    = jpre  + (long)BV * CAUX;             // 32*128
  float* o2    = o1    + CB * CAUX;                   // 32*256
  float* o3    = o2    + CB * CL1;                    // 32*256

  // 1) node -> concat slots
  scatter_node<<<dim3((BV * CMF + 255) / 256), 256, 0, stream>>>(node, ann1, ann2);

  // 2) tmp[e] = node @ W_adj0[e] + b_adj0[e]      (z-batched over E)
  gemm_wmma_f32<<<dim3(CU1 / 16, BV / 16, CE), 32, 0, stream>>>(
      node, CMF, 0, W_adj0, CU1, (long)CMF * CU1, b_adj0, CU1,
      tmp, CU1, (long)BV * CU1, 0, CMF, ACT_NONE);

  // 3) selft = node @ W2_0 + b2_0
  gemm_wmma_f32<<<dim3(CU1 / 16, BV / 16, 1), 32, 0, stream>>>(
      node, CMF, 0, W2_0, CU1, 0, b2_0, 0,
      selft, CU1, 0, 0, CMF, ACT_NONE);

  // 4) h0 = tanh(adj-contract + selft) -> ann1[:, 16:144]   (8 waves/block)
  gcl_adj_wmma<<<dim3(1, CV / 16, CB), (CU1 / 16) * 32, 0, stream>>>(
      adj, tmp, selft, ann1, CMF + CU1, CMF, CU1);

  // 5) tmp[e] = ann1 @ W_adj1[e] + b_adj1[e]
  gemm_wmma_f32<<<dim3(CU2 / 16, BV / 16, CE), 32, 0, stream>>>(
      ann1, CMF + CU1, 0, W_adj1, CU2, (long)(CMF + CU1) * CU2, b_adj1, CU2,
      tmp, CU2, (long)BV * CU2, 0, CMF + CU1, ACT_NONE);

  // 6) selft = ann1 @ W2_1 + b2_1
  gemm_wmma_f32<<<dim3(CU2 / 16, BV / 16, 1), 32, 0, stream>>>(
      ann1, CMF + CU1, 0, W2_1, CU2, 0, b2_1, 0,
      selft, CU2, 0, 0, CMF + CU1, ACT_NONE);

  // 7) h1 -> ann2[:, 0:64]                                  (4 waves/block)
  gcl_adj_wmma<<<dim3(1, CV / 16, CB), (CU2 / 16) * 32, 0, stream>>>(
      adj, tmp, selft, ann2, CU2 + CMF, 0, CU2);

  // 8) ipre = sigmoid(ann2 @ Wi + bi)
  gemm_wmma_f32<<<dim3(CAUX / 16, BV / 16, 1), 32, 0, stream>>>(
      ann2, CU2 + CMF, 0, Wi, CAUX, 0, bi, 0,
      ipre, CAUX, 0, 0, CU2 + CMF, ACT_SIG);

  // 9) jpre = tanh(ann2 @ Wj + bj)
  gemm_wmma_f32<<<dim3(CAUX / 16, BV / 16, 1), 32, 0, stream>>>(
      ann2, CU2 + CMF, 0, Wj, CAUX, 0, bj, 0,
      jpre, CAUX, 0, 0, CU2 + CMF, ACT_TANH);

  // 10) graph pooling gate
  reduce_o<<<dim3(CB), CAUX, 0, stream>>>(ipre, jpre, o1);

  // 11) o2 = tanh(o1 @ Wd1 + bd1)
  gemm_wmma_f32<<<dim3(CL1 / 16, CB / 16, 1), 32, 0, stream>>>(
      o1, CAUX, 0, Wd1, CL1, 0, bd1, 0,
      o2, CL1, 0, 0, CAUX, ACT_TANH);

  // 12) o3 = tanh(o2 @ Wd2 + bd2)
  gemm_wmma_f32<<<dim3(CL2 / 16, CB / 16, 1), 32, 0, stream>>>(
      o2, CL1, 0, Wd2, CL2, 0, bd2, 0,
      o3, CL2, 0, 0, CL1, ACT_TANH);

  // 13) VAE head: mu / logvar / reparameterized output
  final_vae<<<dim3(CEMB / 256), 256, 0, stream>>>(o3, Wmu, bmu, Wlv, blv, eps, out);
}